// HeteroImputeClassifyModel_11398843203883
// MI455X (gfx1250) — compile-verified
//
#include <hip/hip_runtime.h>
#include <hip/hip_bf16.h>
#include <cstddef>

#define HD 128
#define GNUM 64
#define OUTD 10

#define APAD 132                     // padded row stride (132 % 64 = 4 -> bank spread)
#define SMEM_FLOATS (16 * APAD + HD * APAD)
#define SMEM_BYTES  (SMEM_FLOATS * 4)

typedef __attribute__((ext_vector_type(2))) float v2f;
typedef __attribute__((ext_vector_type(8))) float v8f;

// ---------------------------------------------------------------------------
// WMMA f32 GEMM:  D[M,128] (+)= A[M,128] @ W[128,128] (+ bias) ; optional ReLU
// 256 threads = 8 waves; wave w computes rows [m0,m0+16) x cols [16w,16w+16)
// using V_WMMA_F32_16X16X4_F32, K swept in steps of 4 (32 WMMAs per tile).
// Both the 16x128 A tile AND the full 128x128 W are staged in (dynamic) LDS:
// ~76KB/block -> 4 workgroups per 320KB WGP, and the WMMA inner loop touches
// only ds_load (no VMEM waits on the critical path).
// ---------------------------------------------------------------------------
__global__ __launch_bounds__(256) void gemm128_wmma(
    const float* __restrict__ A, const float* __restrict__ W,
    const float* __restrict__ bias, float* __restrict__ D,
    int M, int accum, int relu)
{
    extern __shared__ float smem[];
    float* As = smem;                 // 16  x APAD
    float* Ws = smem + 16 * APAD;     // 128 x APAD

    const int tid  = threadIdx.x;
    const int wave = tid >> 5;
    const int lane = tid & 31;
    const int m0   = blockIdx.x << 4;

    // stage A tile (16x128) as float4, zero-pad rows past M
    for (int i = tid; i < 16 * (HD / 4); i += 256) {
        int r  = i >> 5;              // /32 float4s per row
        int c4 = i & 31;
        int gm = m0 + r;
        float4 v = make_float4(0.f, 0.f, 0.f, 0.f);
        if (gm < M) v = ((const float4*)(A + (size_t)gm * HD))[c4];
        ((float4*)(As + r * APAD))[c4] = v;
    }
    // stage W (128x128) as float4
    for (int i = tid; i < HD * (HD / 4); i += 256) {
        int k  = i >> 5;
        int c4 = i & 31;
        ((float4*)(Ws + k * APAD))[c4] = ((const float4*)(W + (size_t)k * HD))[c4];
    }
    __syncthreads();

    const int n0   = wave << 4;
    const int half = lane >> 4;       // 0: lanes 0-15, 1: lanes 16-31
    const int r    = lane & 15;

    v8f acc = {};
#pragma unroll
    for (int k0 = 0; k0 < HD; k0 += 4) {
        v2f a, b;
        // A 16x4 frag: lane r holds row M=r; VGPR0=K(0|2), VGPR1=K(1|3)
        a.x = As[r * APAD + k0 + 2 * half];
        a.y = As[r * APAD + k0 + 2 * half + 1];
        // B 4x16 frag: VGPR0 = rows K(0|2), VGPR1 = rows K(1|3), N striped on lanes
        b.x = Ws[(k0 + 2 * half) * APAD + n0 + r];
        b.y = Ws[(k0 + 2 * half + 1) * APAD + n0 + r];
        acc = __builtin_amdgcn_wmma_f32_16x16x4_f32(false, a, false, b,
                                                    (short)0, acc, false, false);
    }

    const float bv = bias ? bias[n0 + r] : 0.f;
#pragma unroll
    for (int i = 0; i < 8; ++i) {
        int row = m0 + i + 8 * half;  // C/D: VGPR i -> M = i (+8 for hi half)
        if (row < M) {
            size_t idx = (size_t)row * HD + n0 + r;
            float v = acc[i] + bv;
            if (accum) v += D[idx];
            if (relu)  v = fmaxf(v, 0.f);
            D[idx] = v;
        }
    }
}

// ---- first MLP layer from 2-wide input: H1 = relu(X[N,2] @ W1[2,128] + b1) ----
__global__ void mlp_in2(const float* __restrict__ X, const float* __restrict__ W1,
                        const float* __restrict__ B1, float* __restrict__ H1, int N)
{
    int idx = blockIdx.x * blockDim.x + threadIdx.x;
    int n = idx >> 7, c = idx & 127;
    if (n >= N) return;
    float x0 = X[2 * n], x1 = X[2 * n + 1];
    float v = fmaf(x0, W1[c], fmaf(x1, W1[HD + c], B1[c]));
    H1[(size_t)n * HD + c] = fmaxf(v, 0.f);
}

__global__ void zero_f32(float* __restrict__ p, long long n)
{
    long long i = blockIdx.x * (long long)blockDim.x + threadIdx.x;
    if (i < n) p[i] = 0.f;
}

// ---- edge scatter: agg[dst] += h_src[src], cnt[dst] += 1 (32 lanes/edge) ----
__global__ void edge_scatter(const float* __restrict__ Hsrc, const int* __restrict__ src,
                             const int* __restrict__ dst, float* __restrict__ agg,
                             float* __restrict__ cnt, int E)
{
    long long gid = blockIdx.x * (long long)blockDim.x + threadIdx.x;
    int e = (int)(gid >> 5);
    int part = (int)(gid & 31);
    if (e >= E) return;
    int s = src[e], d = dst[e];
    float4 v = ((const float4*)(Hsrc + (size_t)s * HD))[part];
    float* ap = agg + (size_t)d * HD + part * 4;
    atomicAdd(ap + 0, v.x);
    atomicAdd(ap + 1, v.y);
    atomicAdd(ap + 2, v.z);
    atomicAdd(ap + 3, v.w);
    if (part == 0) atomicAdd(cnt + d, 1.f);
}

// ---- agg /= max(cnt, 1) ----
__global__ void agg_norm(float* __restrict__ agg, const float* __restrict__ cnt, int N)
{
    int idx = blockIdx.x * blockDim.x + threadIdx.x;
    int n = idx >> 7, c = idx & 127;
    if (n >= N) return;
    agg[(size_t)n * HD + c] /= fmaxf(cnt[n], 1.f);
}

// ---- out[n] = H1[n,:] . W2[128,1] + b2 ----
__global__ void dot128(const float* __restrict__ H1, const float* __restrict__ W2,
                       const float* __restrict__ B2, float* __restrict__ out, int N)
{
    int n = blockIdx.x * blockDim.x + threadIdx.x;
    if (n >= N) return;
    const float4* h = (const float4*)(H1 + (size_t)n * HD);
    const float4* w = (const float4*)W2;
    float acc = 0.f;
#pragma unroll
    for (int i = 0; i < 32; ++i) {
        float4 a = h[i], b = w[i];
        acc = fmaf(a.x, b.x, fmaf(a.y, b.y, fmaf(a.z, b.z, fmaf(a.w, b.w, acc))));
    }
    out[n] = acc + B2[0];
}

// ---- fa = [flag==0 ? x0 : pred, flag] ----
__global__ void fill_vals(const float* __restrict__ X, const float* __restrict__ pred,
                          float* __restrict__ F, int N)
{
    int n = blockIdx.x * blockDim.x + threadIdx.x;
    if (n >= N) return;
    float x0 = X[2 * n], flag = X[2 * n + 1];
    F[2 * n]     = (flag == 0.f) ? x0 : pred[n];
    F[2 * n + 1] = flag;
}

// ---- global_add_pool: g[batch[n]] += V[n] ----
__global__ void pool_add(const float* __restrict__ V, const int* __restrict__ batch,
                         float* __restrict__ g, int N)
{
    int idx = blockIdx.x * blockDim.x + threadIdx.x;
    int n = idx >> 7, c = idx & 127;
    if (n >= N) return;
    atomicAdd(g + (size_t)batch[n] * HD + c, V[(size_t)n * HD + c]);
}

// ---- head: emb=[g_a|g_b] (256) -> relu(@W1[256,128]+b1) -> @W2[128,10]+b2 ----
__global__ __launch_bounds__(128) void head_mlp(
    const float* __restrict__ ga, const float* __restrict__ gb,
    const float* __restrict__ W1, const float* __restrict__ B1,
    const float* __restrict__ W2, const float* __restrict__ B2,
    float* __restrict__ logits, float* __restrict__ emb_out)
{
    __shared__ float emb[2 * HD];
    __shared__ float h1[HD];
    const int g = blockIdx.x, c = threadIdx.x;
    emb[c]      = ga[(size_t)g * HD + c];
    emb[HD + c] = gb[(size_t)g * HD + c];
    __syncthreads();
    emb_out[(size_t)g * 2 * HD + c]      = emb[c];
    emb_out[(size_t)g * 2 * HD + HD + c] = emb[HD + c];
    float acc = B1[c];
#pragma unroll 8
    for (int k = 0; k < 2 * HD; ++k) acc = fmaf(emb[k], W1[(size_t)k * HD + c], acc);
    h1[c] = fmaxf(acc, 0.f);
    __syncthreads();
    if (c < OUTD) {
        float a2 = B2[c];
#pragma unroll 8
        for (int k = 0; k < HD; ++k) a2 = fmaf(h1[k], W2[(size_t)k * OUTD + c], a2);
        logits[(size_t)g * OUTD + c] = a2;
    }
}

// ---------------------------------------------------------------------------
static inline int cdiv(long long a, int b) { return (int)((a + b - 1) / b); }

extern "C" void kernel_launch(void* const* d_in, const int* in_sizes, int n_in,
                              void* d_out, int out_size, void* d_ws, size_t ws_size,
                              hipStream_t stream)
{
    // ---- input index map (params pytree: dict keys alphabetical) ----
    // 0..3 dec_a(w1,b1,w2,b2) 4..7 dec_b 8..11 fil_a 12..15 fil_b
    // 16..19 head 20..23 inp_a 24..27 inp_b
    // sage: 28..30 l0.aa(wl,bl,wr) 31..33 l0.ab 34..36 l0.ba
    //       37..39 l1.aa          40..42 l1.ab 43..45 l1.ba
    // 46 x_a 47 x_b 48 src_ab 49 dst_ab 50 src_ba 51 dst_ba 52 src_aa 53 dst_aa
    // 54 batch_a 55 batch_b
    const float* P[46];
    for (int i = 0; i < 46; ++i) P[i] = (const float*)d_in[i];
    const float* x_a = (const float*)d_in[46];
    const float* x_b = (const float*)d_in[47];
    const int* src_ab = (const int*)d_in[48];
    const int* dst_ab = (const int*)d_in[49];
    const int* src_ba = (const int*)d_in[50];
    const int* dst_ba = (const int*)d_in[51];
    const int* src_aa = (const int*)d_in[52];
    const int* dst_aa = (const int*)d_in[53];
    const int* batch_a = (const int*)d_in[54];
    const int* batch_b = (const int*)d_in[55];

    const int NA = in_sizes[46] / 2;
    const int NB = in_sizes[47] / 2;
    const int E_ab = in_sizes[48], E_ba = in_sizes[50], E_aa = in_sizes[52];
    const int NM = (NA > NB) ? NA : NB;

    // ---- workspace layout (floats) ----
    float* w = (float*)d_ws;
    float* hA  = w;                 w += (size_t)NA * HD;
    float* hB  = w;                 w += (size_t)NB * HD;
    float* hA2 = w;                 w += (size_t)NA * HD;
    float* hB2 = w;                 w += (size_t)NB * HD;
    float* agg = w;                 w += (size_t)NM * HD;
    float* cnt = w;                 w += (size_t)NM;
    float* tmp = w;                 w += (size_t)NM * HD;   // MLP hidden / fil output
    float* gA  = w;                 w += (size_t)GNUM * HD;
    float* gB  = w;                 w += (size_t)GNUM * HD;

    // ---- output layout ----
    float* out = (float*)d_out;
    float* o_logits = out;
    float* o_pa  = out + (size_t)GNUM * OUTD;
    float* o_pb  = o_pa + NA;
    float* o_fa  = o_pb + NB;
    float* o_fb  = o_fa + (size_t)2 * NA;
    float* o_ha  = o_fb + (size_t)2 * NB;
    float* o_hb  = o_ha + (size_t)NA * HD;
    float* o_emb = o_hb + (size_t)NB * HD;

    const dim3 blk(256);
    #define GEMM(Aptr, Wptr, Bptr, Dptr, M, acc, rel) \
        gemm128_wmma<<<cdiv((long long)(M), 16), blk, SMEM_BYTES, stream>>>(Aptr, Wptr, Bptr, Dptr, M, acc, rel)

    // ---- encode ----
    mlp_in2<<<cdiv((long long)NA * HD, 256), blk, 0, stream>>>(x_a, P[20], P[21], tmp, NA);
    GEMM(tmp, P[22], P[23], hA, NA, 0, 0);
    mlp_in2<<<cdiv((long long)NB * HD, 256), blk, 0, stream>>>(x_b, P[24], P[25], tmp, NB);
    GEMM(tmp, P[26], P[27], hB, NB, 0, 0);

    // ---- SAGE layers ----
    for (int l = 0; l < 2; ++l) {
        const float* aa_wl = P[28 + 9 * l + 0]; const float* aa_bl = P[28 + 9 * l + 1];
        const float* aa_wr = P[28 + 9 * l + 2];
        const float* ab_wl = P[28 + 9 * l + 3]; const float* ab_bl = P[28 + 9 * l + 4];
        const float* ab_wr = P[28 + 9 * l + 5];
        const float* ba_wl = P[28 + 9 * l + 6]; const float* ba_bl = P[28 + 9 * l + 7];
        const float* ba_wr = P[28 + 9 * l + 8];

        // a->b  (dst type b): h_b' = relu(mean_ab@wl + bl + h_b@wr)
        zero_f32<<<cdiv((long long)NB * HD, 256), blk, 0, stream>>>(agg, (long long)NB * HD);
        zero_f32<<<cdiv((long long)NB, 256), blk, 0, stream>>>(cnt, NB);
        edge_scatter<<<cdiv((long long)E_ab * 32, 256), blk, 0, stream>>>(hA, src_ab, dst_ab, agg, cnt, E_ab);
        agg_norm<<<cdiv((long long)NB * HD, 256), blk, 0, stream>>>(agg, cnt, NB);
        GEMM(agg, ab_wl, ab_bl, hB2, NB, 0, 0);
        GEMM(hB,  ab_wr, nullptr, hB2, NB, 1, 1);

        // b->a
        zero_f32<<<cdiv((long long)NA * HD, 256), blk, 0, stream>>>(agg, (long long)NA * HD);
        zero_f32<<<cdiv((long long)NA, 256), blk, 0, stream>>>(cnt, NA);
        edge_scatter<<<cdiv((long long)E_ba * 32, 256), blk, 0, stream>>>(hB, src_ba, dst_ba, agg, cnt, E_ba);
        agg_norm<<<cdiv((long long)NA * HD, 256), blk, 0, stream>>>(agg, cnt, NA);
        GEMM(agg, ba_wl, ba_bl, hA2, NA, 0, 0);
        GEMM(hA,  ba_wr, nullptr, hA2, NA, 1, 0);

        // a->a (sum into hA2), relu on the last accumulate
        zero_f32<<<cdiv((long long)NA * HD, 256), blk, 0, stream>>>(agg, (long long)NA * HD);
        zero_f32<<<cdiv((long long)NA, 256), blk, 0, stream>>>(cnt, NA);
        edge_scatter<<<cdiv((long long)E_aa * 32, 256), blk, 0, stream>>>(hA, src_aa, dst_aa, agg, cnt, E_aa);
        agg_norm<<<cdiv((long long)NA * HD, 256), blk, 0, stream>>>(agg, cnt, NA);
        GEMM(agg, aa_wl, aa_bl, hA2, NA, 1, 0);
        GEMM(hA,  aa_wr, nullptr, hA2, NA, 1, 1);

        float* t;
        t = hA; hA = hA2; hA2 = t;
        t = hB; hB = hB2; hB2 = t;
    }

    // ---- export final hidden states ----
    hipMemcpyAsync(o_ha, hA, (size_t)NA * HD * sizeof(float), hipMemcpyDeviceToDevice, stream);
    hipMemcpyAsync(o_hb, hB, (size_t)NB * HD * sizeof(float), hipMemcpyDeviceToDevice, stream);

    // ---- decode ----
    GEMM(hA, P[0], P[1], tmp, NA, 0, 1);
    dot128<<<cdiv((long long)NA, 256), blk, 0, stream>>>(tmp, P[2], P[3], o_pa, NA);
    GEMM(hB, P[4], P[5], tmp, NB, 0, 1);
    dot128<<<cdiv((long long)NB, 256), blk, 0, stream>>>(tmp, P[6], P[7], o_pb, NB);

    // ---- fill missing ----
    fill_vals<<<cdiv((long long)NA, 256), blk, 0, stream>>>(x_a, o_pa, o_fa, NA);
    fill_vals<<<cdiv((long long)NB, 256), blk, 0, stream>>>(x_b, o_pb, o_fb, NB);

    // ---- fill MLP + pooling ----
    zero_f32<<<cdiv((long long)2 * GNUM * HD, 256), blk, 0, stream>>>(gA, (long long)2 * GNUM * HD); // gA,gB contiguous
    mlp_in2<<<cdiv((long long)NA * HD, 256), blk, 0, stream>>>(o_fa, P[8], P[9], tmp, NA);
    GEMM(tmp, P[10], P[11], agg, NA, 0, 0);
    pool_add<<<cdiv((long long)NA * HD, 256), blk, 0, stream>>>(agg, batch_a, gA, NA);
    mlp_in2<<<cdiv((long long)NB * HD, 256), blk, 0, stream>>>(o_fb, P[12], P[13], tmp, NB);
    GEMM(tmp, P[14], P[15], agg, NB, 0, 0);
    pool_add<<<cdiv((long long)NB * HD, 256), blk, 0, stream>>>(agg, batch_b, gB, NB);

    // ---- head ----
    head_mlp<<<GNUM, 128, 0, stream>>>(gA, gB, P[16], P[17], P[18], P[19], o_logits, o_emb);
    #undef GEMM
}